// LGMLoss_v0_90013924589654
// MI455X (gfx1250) — compile-verified
//
#include <hip/hip_runtime.h>

#define B_SZ 4096
#define C_SZ 10000
#define D_SZ 256
#define ALPHA 0.1f

typedef float v2f __attribute__((ext_vector_type(2)));
typedef float v4f __attribute__((ext_vector_type(4)));
typedef float v8f __attribute__((ext_vector_type(8)));

// ---------------------------------------------------------------------------
// Wave-per-row sum of squares: out[r] = sum_k x[r][k]^2   (D_SZ = 256)
// ---------------------------------------------------------------------------
__global__ __launch_bounds__(256) void lgm_rowsq(const float* __restrict__ x,
                                                 float* __restrict__ out,
                                                 int nrows) {
  int row  = (int)((blockIdx.x * blockDim.x + threadIdx.x) >> 5);
  int lane = threadIdx.x & 31;
  if (row >= nrows) return;
  const float* r = x + (size_t)row * D_SZ;
  float s = 0.f;
#pragma unroll
  for (int i = 0; i < D_SZ / 32; ++i) {
    float v = r[lane + i * 32];
    s += v * v;
  }
#pragma unroll
  for (int off = 16; off > 0; off >>= 1) s += __shfl_xor(s, off, 32);
  if (lane == 0) out[row] = s;
}

// ---------------------------------------------------------------------------
// Zero the likelihood slot (d_out is poisoned by the harness).
// ---------------------------------------------------------------------------
__global__ void lgm_zero(float* p) {
  if (threadIdx.x == 0 && blockIdx.x == 0) p[0] = 0.f;
}

// ---------------------------------------------------------------------------
// likelihood = sum_b ||feat[b] - centers[label[b]]||^2 / (2B)
// wave-per-row, shuffle reduce, atomicAdd of the scaled partial.
// ---------------------------------------------------------------------------
__global__ __launch_bounds__(256) void lgm_lik(const float* __restrict__ feat,
                                               const long long* __restrict__ label,
                                               const float* __restrict__ centers,
                                               float* __restrict__ out_lik) {
  int row  = (int)((blockIdx.x * blockDim.x + threadIdx.x) >> 5);
  int lane = threadIdx.x & 31;
  if (row >= B_SZ) return;
  const float* f = feat + (size_t)row * D_SZ;
  const float* c = centers + (size_t)label[row] * D_SZ;
  float s = 0.f;
#pragma unroll
  for (int i = 0; i < D_SZ / 32; ++i) {
    float d = f[lane + i * 32] - c[lane + i * 32];
    s += d * d;
  }
#pragma unroll
  for (int off = 16; off > 0; off >>= 1) s += __shfl_xor(s, off, 32);
  if (lane == 0) atomicAdd(out_lik, s * (1.0f / (2.0f * (float)B_SZ)));
}

// ---------------------------------------------------------------------------
// Fused GEMM + epilogue:
//   cross = feat @ centers^T         (WMMA f32 16x16x4)
//   dist  = fsq[r] + csq[c] - 2*cross
//   logits        = -0.5 * dist
//   margin_logits = -0.5 * dist * (1 + alpha*[c == label[r]])
//
// Block tile 128x128, 8 waves, wave tile 64x32 (4x2 accs of v8f).
// K staged through LDS in 32-wide chunks, rows padded to 36 floats
// (stride 36: lane bank = (36*lane + k) mod 64 covers all 64 banks
// across the two half-waves -> conflict free v2f fragment reads).
// ---------------------------------------------------------------------------
__global__ __launch_bounds__(256) void lgm_gemm(const float* __restrict__ feat,
                                                const long long* __restrict__ label,
                                                const float* __restrict__ centers,
                                                const float* __restrict__ fsq,
                                                const float* __restrict__ csq,
                                                float* __restrict__ out) {
  __shared__ float ldsA[128 * 36];
  __shared__ float ldsB[128 * 36];
  __shared__ int   ldsLab[128];
  __shared__ float ldsFsq[128];

  const int tid   = threadIdx.x;
  const int lane  = tid & 31;
  const int wave  = tid >> 5;
  const int waveM = wave >> 2;  // 0..1  -> 64 rows each
  const int waveN = wave & 3;   // 0..3  -> 32 cols each
  const int rowBase = blockIdx.y * 128;
  const int colBase = blockIdx.x * 128;

  if (tid < 128) {
    ldsLab[tid] = (int)label[rowBase + tid];
    ldsFsq[tid] = fsq[rowBase + tid];
  }

  v8f acc[4][2];
#pragma unroll
  for (int mi = 0; mi < 4; ++mi)
#pragma unroll
    for (int ni = 0; ni < 2; ++ni) acc[mi][ni] = (v8f){};

  const int rowHalf = lane & 15;       // N index within 16, or M row within 16
  const int khalf   = (lane >> 4) * 2; // K sub-offset for the upper half-wave

  for (int kc = 0; kc < D_SZ; kc += 32) {
    __syncthreads();
    // Stage A (feat tile) and B (centers tile): 128 rows x 32 floats each.
    // 256 threads x 4 groups; each group is one float4 of one row.
#pragma unroll
    for (int i = 0; i < 4; ++i) {
      int g  = tid + i * 256;
      int r  = g >> 3;  // 0..127
      int c4 = g & 7;   // 0..7  -> float4 slot
      v4f va = *(const v4f*)&feat[(size_t)(rowBase + r) * D_SZ + kc + c4 * 4];
      *(v4f*)&ldsA[r * 36 + c4 * 4] = va;
      int cls = colBase + r;
      v4f vb = (v4f){};
      if (cls < C_SZ) vb = *(const v4f*)&centers[(size_t)cls * D_SZ + kc + c4 * 4];
      *(v4f*)&ldsB[r * 36 + c4 * 4] = vb;
      // Prefetch next K chunk (gfx1250 global_prefetch_b8 path).
      if (kc + 32 < D_SZ) {
        __builtin_prefetch(&feat[(size_t)(rowBase + r) * D_SZ + kc + 32 + c4 * 4], 0, 0);
        if (cls < C_SZ)
          __builtin_prefetch(&centers[(size_t)cls * D_SZ + kc + 32 + c4 * 4], 0, 0);
      }
    }
    __syncthreads();

#pragma unroll
    for (int kk = 0; kk < 32; kk += 4) {
      v2f afrag[4];
#pragma unroll
      for (int mi = 0; mi < 4; ++mi) {
        int r = waveM * 64 + mi * 16 + rowHalf;
        afrag[mi] = *(const v2f*)&ldsA[r * 36 + kk + khalf];
      }
      v2f bfrag[2];
#pragma unroll
      for (int ni = 0; ni < 2; ++ni) {
        int c = waveN * 32 + ni * 16 + rowHalf;
        bfrag[ni] = *(const v2f*)&ldsB[c * 36 + kk + khalf];
      }
#pragma unroll
      for (int mi = 0; mi < 4; ++mi)
#pragma unroll
        for (int ni = 0; ni < 2; ++ni)
          acc[mi][ni] = __builtin_amdgcn_wmma_f32_16x16x4_f32(
              false, afrag[mi], false, bfrag[ni], (short)0, acc[mi][ni],
              false, false);
    }
  }

  // Epilogue: C/D layout -> lanes 0-15: M = v, lanes 16-31: M = 8+v; N = lane&15.
  const size_t mOff = (size_t)B_SZ * C_SZ;
#pragma unroll
  for (int mi = 0; mi < 4; ++mi) {
#pragma unroll
    for (int ni = 0; ni < 2; ++ni) {
      int col = colBase + waveN * 32 + ni * 16 + rowHalf;
      if (col < C_SZ) {
        float cs = csq[col];
#pragma unroll
        for (int v = 0; v < 8; ++v) {
          int rl = waveM * 64 + mi * 16 + (lane >> 4) * 8 + v;
          int gr = rowBase + rl;
          float dist = ldsFsq[rl] + cs - 2.0f * acc[mi][ni][v];
          float lg = -0.5f * dist;
          size_t idx = (size_t)gr * C_SZ + col;
          out[idx] = lg;
          float w = (ldsLab[rl] == col) ? (1.0f + ALPHA) : 1.0f;
          out[mOff + idx] = lg * w;
        }
      }
    }
  }
}

// ---------------------------------------------------------------------------
extern "C" void kernel_launch(void* const* d_in, const int* in_sizes, int n_in,
                              void* d_out, int out_size, void* d_ws, size_t ws_size,
                              hipStream_t stream) {
  const float*     feat    = (const float*)d_in[0];
  const long long* label   = (const long long*)d_in[1];  // int64 per reference
  const float*     centers = (const float*)d_in[2];
  float* out = (float*)d_out;

  float* fsq = (float*)d_ws;          // [B_SZ]
  float* csq = fsq + B_SZ;            // [C_SZ]
  float* lik = out + (size_t)2 * B_SZ * C_SZ;

  lgm_rowsq<<<B_SZ / 8, 256, 0, stream>>>(feat, fsq, B_SZ);
  lgm_rowsq<<<(C_SZ + 7) / 8, 256, 0, stream>>>(centers, csq, C_SZ);
  lgm_zero<<<1, 32, 0, stream>>>(lik);
  lgm_lik<<<B_SZ / 8, 256, 0, stream>>>(feat, label, centers, lik);

  dim3 grid((C_SZ + 127) / 128, B_SZ / 128);  // 79 x 32 blocks
  lgm_gemm<<<grid, 256, 0, stream>>>(feat, label, centers, fsq, csq, out);
}